// VanillaGNNLayer_21904333209666
// MI455X (gfx1250) — compile-verified
//
#include <hip/hip_runtime.h>

typedef __attribute__((ext_vector_type(2))) float v2f;
typedef __attribute__((ext_vector_type(8))) float v8f;

#define DIM 128   // DIM_IN == DIM_OUT == 128

// ---------------------------------------------------------------------------
// Kernel 1: h = x @ W^T using V_WMMA_F32_16X16X4_F32.
// Block = 256 threads = 8 waves. Block b owns node rows [16b, 16b+16).
// Wave w owns output columns [16w, 16w+16). K loop: 128/4 = 32 WMMA steps.
//
// A (16x4 f32) lane layout: M = lane%16, VGPR r holds K = 2*(lane/16) + r
//   -> lane loads float2 x[row][kb + 2*(lane>>4)]
// B (4x16 f32) lane layout (B = W^T): N = lane%16, same K striping
//   -> lane loads float2 W[col][kb + 2*(lane>>4)]
// C/D (16x16 f32): VGPR v -> M = v + 8*(lane>>4), N = lane%16
// ---------------------------------------------------------------------------
__global__ __launch_bounds__(256) void gnn_gemm_wmma(
    const float* __restrict__ x, const float* __restrict__ W,
    float* __restrict__ h, int n_nodes)
{
    const int lane  = threadIdx.x & 31;
    const int wave  = threadIdx.x >> 5;   // 0..7 : which 16-wide column tile
    const int m     = lane & 15;
    const int khalf = lane >> 4;          // 0 or 1

    int row = blockIdx.x * 16 + m;        // node row this lane feeds into A
    if (row >= n_nodes) row = n_nodes - 1;   // clamp loads, keep EXEC all-1s
    const int col = wave * 16 + m;        // output dim this lane feeds into B

    const float* xrow = x + (size_t)row * DIM + 2 * khalf;
    const float* wrow = W + (size_t)col * DIM + 2 * khalf;

    v8f c = {};
#pragma unroll
    for (int kb = 0; kb < DIM; kb += 4) {
        v2f a = *reinterpret_cast<const v2f*>(xrow + kb);
        v2f b = *reinterpret_cast<const v2f*>(wrow + kb);
        // (neg_a, A, neg_b, B, c_mod, C, reuse_a, reuse_b)
        c = __builtin_amdgcn_wmma_f32_16x16x4_f32(
                false, a, false, b, (short)0, c, false, false);
    }

    const int n_out = wave * 16 + (lane & 15);
#pragma unroll
    for (int v = 0; v < 8; ++v) {
        const int m_out  = v + 8 * khalf;
        const int row_out = blockIdx.x * 16 + m_out;
        if (row_out < n_nodes)
            h[(size_t)row_out * DIM + n_out] = c[v];
    }
}

// ---------------------------------------------------------------------------
// Kernel 2: out[dst[e]] += vals[e] * h[src[e]]
// One wave per edge; lane l handles dims [4l, 4l+4) -> float4 gather +
// 4 hardware global_atomic_add_f32. h and out both fit in the 192 MB L2,
// so gathers and atomics resolve on-die.
// ---------------------------------------------------------------------------
__global__ __launch_bounds__(256) void gnn_edge_scatter(
    const float* __restrict__ h, const float* __restrict__ vals,
    const int* __restrict__ src, const int* __restrict__ dst,
    float* __restrict__ out, int n_edges)
{
    const int lane = threadIdx.x & 31;
    const int e    = blockIdx.x * 8 + (threadIdx.x >> 5); // 8 waves/block
    if (e >= n_edges) return;

    const float v = vals[e];
    const int   s = src[e];
    const int   d = dst[e];

    const float4 hv = *reinterpret_cast<const float4*>(
        h + (size_t)s * DIM + lane * 4);
    float* o = out + (size_t)d * DIM + lane * 4;

    unsafeAtomicAdd(o + 0, v * hv.x);
    unsafeAtomicAdd(o + 1, v * hv.y);
    unsafeAtomicAdd(o + 2, v * hv.z);
    unsafeAtomicAdd(o + 3, v * hv.w);
}

// ---------------------------------------------------------------------------
// Launch: zero out, GEMM into workspace h, then edge scatter.
// Inputs (setup_inputs order): x[100000*128] f32, W[128*128] f32,
// vals[1.6M] f32, src[1.6M] i32, dst[1.6M] i32. out: 100000*128 f32.
// Workspace: h = 100000*128*4 = 51.2 MB in d_ws.
// ---------------------------------------------------------------------------
extern "C" void kernel_launch(void* const* d_in, const int* in_sizes, int n_in,
                              void* d_out, int out_size, void* d_ws, size_t ws_size,
                              hipStream_t stream)
{
    const float* x    = (const float*)d_in[0];
    const float* W    = (const float*)d_in[1];
    const float* vals = (const float*)d_in[2];
    const int*   src  = (const int*)d_in[3];
    const int*   dst  = (const int*)d_in[4];
    float*       out  = (float*)d_out;

    const int n_nodes = in_sizes[0] / DIM;  // 100000
    const int n_edges = in_sizes[2];        // 1600000

    float* h = (float*)d_ws;                // n_nodes*DIM f32 scratch

    hipMemsetAsync(d_out, 0, (size_t)out_size * sizeof(float), stream);

    const int row_tiles = (n_nodes + 15) / 16;
    gnn_gemm_wmma<<<dim3(row_tiles), dim3(256), 0, stream>>>(x, W, h, n_nodes);

    const int blocks = (n_edges + 7) / 8;   // 8 edges (waves) per 256-thr block
    gnn_edge_scatter<<<dim3(blocks), dim3(256), 0, stream>>>(
        h, vals, src, dst, out, n_edges);
}